// SAM1D_46892452937836
// MI455X (gfx1250) — compile-verified
//
#include <hip/hip_runtime.h>
#include <math.h>

// Problem constants (from reference)
#define BQ 16
#define CQ 512
#define LQ 8192
#define KQ 7
#define TL 64           // L-positions per tile/block
#define NTHREADS 256    // 8 waves of 32

typedef int   i32x4 __attribute__((ext_vector_type(4)));
typedef float f32x4 __attribute__((ext_vector_type(4)));
typedef __attribute__((address_space(1))) i32x4* gbl_v4i_ptr;   // global
typedef __attribute__((address_space(3))) i32x4* lds_v4i_ptr;   // LDS

// ---- CDNA5 async global->LDS copy (16 bytes per lane) ----------------------
__device__ __forceinline__ void async_copy_f4(const float* gsrc, float* ldst) {
#if defined(__gfx1250__) && __has_builtin(__builtin_amdgcn_global_load_async_to_lds_b128)
  // Signature (probe-confirmed): (int4 AS1*, int4 AS3*, imm offset, imm cpol)
  __builtin_amdgcn_global_load_async_to_lds_b128(
      (gbl_v4i_ptr)(unsigned long long)gsrc,
      (lds_v4i_ptr)(unsigned)(unsigned long long)ldst,
      /*offset=*/0, /*cpol=*/0);
#else
  asm volatile("global_load_async_to_lds_b128 %0, %1, off"
               :
               : "v"((unsigned)(unsigned long long)ldst), "v"(gsrc)
               : "memory");
#endif
}

__device__ __forceinline__ void wait_async0() {
#if defined(__gfx1250__) && __has_builtin(__builtin_amdgcn_s_wait_asynccnt)
  __builtin_amdgcn_s_wait_asynccnt(0);
#else
  asm volatile("s_wait_asynccnt 0" ::: "memory");
#endif
}

// ---------------------------------------------------------------------------
__global__ __launch_bounds__(NTHREADS) void sam1d_fused_kernel(
    const float* __restrict__ x,     // (B, C, L)
    const float* __restrict__ w,     // (1, 2, 7): w[0..6]=max-ch, w[7..13]=avg-ch
    float* __restrict__ out)         // (B, C, L)
{
  __shared__ float xs[CQ * TL];      // 128 KB: tile, layout [c][l], l stride 64
  __shared__ float pmax[4 * TL];
  __shared__ float psum[4 * TL];
  __shared__ float fm[TL + 6];       // max-pool feats, halo of 3 each side
  __shared__ float fa[TL + 6];       // avg-pool feats
  __shared__ float attn[TL];

  const int tid        = threadIdx.x;
  const int tilesPerB  = LQ / TL;                    // 128
  const int b          = blockIdx.x / tilesPerB;
  const int l0         = (blockIdx.x - b * tilesPerB) * TL;

  const float* xb = x + ((size_t)b * CQ) * LQ + l0;  // tile base (c=0, l=l0)

  // ---- Stage C x TL tile into LDS with async DMA loads --------------------
  // 32768 floats / (256 thr * 4 floats) = 32 b128 async ops per thread.
  #pragma unroll 8
  for (int it = 0; it < (CQ * TL) / (NTHREADS * 4); ++it) {
    const int f = it * (NTHREADS * 4) + tid * 4;     // flat float index
    const int c = f >> 6;                            // TL == 64
    const int l = f & (TL - 1);                      // multiple of 4
    async_copy_f4(xb + (size_t)c * LQ + l, &xs[f]);
  }

  // ---- Halo partials: issue the (strided, high-latency) global loads NOW --
  // so they overlap with the 128 KB async DMA. Independent of LDS contents.
  // Threads 64..255 form 6 wave32 groups, one per halo position.
  float hmx = -__builtin_inff(), hsm = 0.f;
  bool hinb = false;
  if (tid >= TL) {
    const int hid  = (tid - TL) >> 5;                // 0..5
    const int lane = (tid - TL) & 31;
    const int hpos = (hid < 3) ? (l0 - 3 + hid) : (l0 + TL + (hid - 3));
    hinb = (hpos >= 0) && (hpos < LQ);
    if (hinb) {
      const float* xp = x + ((size_t)b * CQ) * LQ + hpos;
      #pragma unroll
      for (int i = 0; i < CQ / 32; ++i) {            // 16 strided channel loads
        const float v = xp[(size_t)(lane + 32 * i) * LQ];
        hmx = fmaxf(hmx, v);
        hsm += v;
      }
    }
  }

  wait_async0();
  __syncthreads();

  // ---- Pass 1: per-group channel reduction (max + sum) from LDS -----------
  {
    const int l  = tid & (TL - 1);
    const int g  = tid >> 6;                         // 0..3 -> 128 channels each
    const int c0 = g * (CQ / 4);
    float mx = -__builtin_inff(), sm = 0.f;
    #pragma unroll 4
    for (int c = c0; c < c0 + CQ / 4; ++c) {
      const float v = xs[c * TL + l];
      mx = fmaxf(mx, v);
      sm += v;
    }
    pmax[g * TL + l] = mx;
    psum[g * TL + l] = sm;
  }
  __syncthreads();

  if (tid < TL) {
    // ---- Combine 4 group partials -> feats for in-tile positions ----------
    float mx = pmax[tid];
    float sm = psum[tid];
    #pragma unroll
    for (int g = 1; g < 4; ++g) {
      mx = fmaxf(mx, pmax[g * TL + tid]);
      sm += psum[g * TL + tid];
    }
    fm[3 + tid] = mx;
    fa[3 + tid] = sm * (1.0f / CQ);
  } else {
    // ---- Finish halo: wave32 tree reduce of the prefetched partials -------
    const int hid  = (tid - TL) >> 5;
    const int lane = (tid - TL) & 31;
    #pragma unroll
    for (int off = 16; off > 0; off >>= 1) {
      hmx = fmaxf(hmx, __shfl_xor(hmx, off, 32));
      hsm += __shfl_xor(hsm, off, 32);
    }
    if (lane == 0) {
      const int p = (hid < 3) ? hid : (3 + TL + (hid - 3));
      fm[p] = hinb ? hmx : 0.f;                      // zero padding of feats
      fa[p] = hinb ? hsm * (1.0f / CQ) : 0.f;
    }
  }
  __syncthreads();

  // ---- 7-tap conv (cross-correlation) + sigmoid ---------------------------
  if (tid < TL) {
    float logit = 0.f;
    #pragma unroll
    for (int j = 0; j < KQ; ++j) {
      logit += w[j] * fm[tid + j] + w[KQ + j] * fa[tid + j];
    }
    attn[tid] = 1.0f / (1.0f + __expf(-logit));
  }
  __syncthreads();

  // ---- Pass 2: out = attn * x, tile read from LDS, NT f4 streaming stores -
  float* ob = out + ((size_t)b * CQ) * LQ + l0;
  #pragma unroll 8
  for (int it = 0; it < (CQ * TL) / (NTHREADS * 4); ++it) {
    const int f = it * (NTHREADS * 4) + tid * 4;
    const int c = f >> 6;
    const int l = f & (TL - 1);
    const f32x4 v = *reinterpret_cast<const f32x4*>(&xs[f]);
    f32x4 r;
    r.x = v.x * attn[l + 0];
    r.y = v.y * attn[l + 1];
    r.z = v.z * attn[l + 2];
    r.w = v.w * attn[l + 3];
    __builtin_nontemporal_store(r, reinterpret_cast<f32x4*>(ob + (size_t)c * LQ + l));
  }
}

// ---------------------------------------------------------------------------
extern "C" void kernel_launch(void* const* d_in, const int* in_sizes, int n_in,
                              void* d_out, int out_size, void* d_ws, size_t ws_size,
                              hipStream_t stream) {
  (void)in_sizes; (void)n_in; (void)out_size; (void)d_ws; (void)ws_size;
  const float* x = (const float*)d_in[0];
  // d_in[1] = mask: intentionally unused (reference's masked_fill results are
  // discarded, so the mask has no effect on the computation).
  const float* w = (const float*)d_in[2];   // 14 floats: (1,2,7)
  float* out = (float*)d_out;

  dim3 grid(BQ * (LQ / TL));   // 16 * 128 = 2048 blocks
  dim3 block(NTHREADS);
  sam1d_fused_kernel<<<grid, block, 0, stream>>>(x, w, out);
}